// LightGCN_75917841924378
// MI455X (gfx1250) — compile-verified
//
#include <hip/hip_runtime.h>

#define N_USERS 100000
#define N_ITEMS 50000
#define NNODES  (N_USERS + N_ITEMS)
#define DIM     64
#define BATCH   8192
#define NLAYERS 3

typedef __attribute__((ext_vector_type(2))) float v2f;
typedef __attribute__((ext_vector_type(8))) float v8f;

// ---------------------------------------------------------------------------
// Init: acc = x_cur = concat(user_emb, item_emb); x_next = 0.   float4 lanes.
// ---------------------------------------------------------------------------
__global__ void lgcn_init(const float* __restrict__ uemb,
                          const float* __restrict__ iemb,
                          float* __restrict__ acc,
                          float* __restrict__ bufA,
                          float* __restrict__ bufB) {
    long idx = (long)blockIdx.x * blockDim.x + threadIdx.x;      // float4 index
    const long tot = (long)NNODES * DIM / 4;
    if (idx >= tot) return;
    const long uq = (long)N_USERS * DIM / 4;
    float4 v = (idx < uq) ? ((const float4*)uemb)[idx]
                          : ((const float4*)iemb)[idx - uq];
    ((float4*)acc)[idx]  = v;
    ((float4*)bufA)[idx] = v;
    ((float4*)bufB)[idx] = make_float4(0.f, 0.f, 0.f, 0.f);
}

// ---------------------------------------------------------------------------
// SpMM scatter: one wave32 per edge. Lane owns 2 of the 64 features.
// y[row] += val * x[col]   via native global_atomic_add_f32 (L2-resident).
// ---------------------------------------------------------------------------
__global__ void lgcn_spmm(const int*   __restrict__ rows,
                          const int*   __restrict__ cols,
                          const float* __restrict__ vals,
                          const float* __restrict__ x,
                          float*       __restrict__ y,
                          int nnz) {
    int wave = blockIdx.x * (blockDim.x >> 5) + (threadIdx.x >> 5);
    int lane = threadIdx.x & 31;
    if (wave >= nnz) return;
    int   r = rows[wave];
    int   c = cols[wave];
    float v = vals[wave];
    float2 xv = ((const float2*)(x + (long)c * DIM))[lane];
    float* yr = y + (long)r * DIM + 2 * lane;
    unsafeAtomicAdd(yr,     v * xv.x);
    unsafeAtomicAdd(yr + 1, v * xv.y);
}

// ---------------------------------------------------------------------------
// acc += xnew, and zero the buffer that becomes next layer's scatter target.
// ---------------------------------------------------------------------------
__global__ void lgcn_accum(float*       __restrict__ acc,
                           const float* __restrict__ xnew,
                           float*       __restrict__ tozero,
                           int do_zero) {
    long idx = (long)blockIdx.x * blockDim.x + threadIdx.x;      // float4 index
    const long tot = (long)NNODES * DIM / 4;
    if (idx >= tot) return;
    float4 a = ((float4*)acc)[idx];
    float4 b = ((const float4*)xnew)[idx];
    a.x += b.x; a.y += b.y; a.z += b.z; a.w += b.w;
    ((float4*)acc)[idx] = a;
    if (do_zero) ((float4*)tozero)[idx] = make_float4(0.f, 0.f, 0.f, 0.f);
}

// ---------------------------------------------------------------------------
// Scoring via V_WMMA_F32_16X16X4_F32: one wave per 16-entry batch tile.
// D = U_tile(16x64) * P_tile^T(64x16) accumulated over 16 K=4 steps; the
// diagonal of D holds the 16 dot products. A/B lane addressing is identical
// for this use: lane L holds row (L%16), k = 4*kk + 2*(L/16) + {0,1}.
// scores scale by (1/4)*(1/4) = 1/16 for light_out = acc/(NLAYERS+1).
// ---------------------------------------------------------------------------
__device__ __forceinline__ float pick8(v8f c, int j) {
    float r = c[0];
    if (j == 1) r = c[1];
    if (j == 2) r = c[2];
    if (j == 3) r = c[3];
    if (j == 4) r = c[4];
    if (j == 5) r = c[5];
    if (j == 6) r = c[6];
    if (j == 7) r = c[7];
    return r;
}

__global__ void lgcn_score(const float* __restrict__ acc,
                           const int*   __restrict__ users,
                           const int*   __restrict__ pos,
                           const int*   __restrict__ neg,
                           float*       __restrict__ pos_scores,
                           float*       __restrict__ neg_scores) {
    int tile = blockIdx.x * (blockDim.x >> 5) + (threadIdx.x >> 5);
    int lane = threadIdx.x & 31;
    int m     = lane & 15;
    int khalf = lane >> 4;                       // 0 or 1
    int b  = tile * 16 + m;
    int u  = users[b];
    int pi = pos[b] + N_USERS;
    int ni = neg[b] + N_USERS;
    const float* urow = acc + (long)u  * DIM;
    const float* prow = acc + (long)pi * DIM;
    const float* nrow = acc + (long)ni * DIM;

    v8f cpos = {};
    v8f cneg = {};
#pragma unroll
    for (int kk = 0; kk < 16; ++kk) {
        int k0 = kk * 4 + khalf * 2;
        float2 ua = *(const float2*)(urow + k0);
        float2 pb = *(const float2*)(prow + k0);
        float2 nb = *(const float2*)(nrow + k0);
        v2f a;  a[0]  = ua.x; a[1]  = ua.y;
        v2f bp; bp[0] = pb.x; bp[1] = pb.y;
        v2f bn; bn[0] = nb.x; bn[1] = nb.y;
        cpos = __builtin_amdgcn_wmma_f32_16x16x4_f32(false, a, false, bp,
                                                     (short)0, cpos, false, false);
        cneg = __builtin_amdgcn_wmma_f32_16x16x4_f32(false, a, false, bn,
                                                     (short)0, cneg, false, false);
    }
    const float s = 1.0f / 16.0f;
    // Diagonal extraction from the 16x16 F32 C/D layout:
    //  m in 0..7  -> VGPR m,   lane m
    //  m in 8..15 -> VGPR m-8, lane m+16
    if (lane < 8) {
        pos_scores[tile * 16 + lane] = pick8(cpos, lane) * s;
        neg_scores[tile * 16 + lane] = pick8(cneg, lane) * s;
    } else if (lane >= 24) {
        int mm = lane - 16;
        pos_scores[tile * 16 + mm] = pick8(cpos, lane - 24) * s;
        neg_scores[tile * 16 + mm] = pick8(cneg, lane - 24) * s;
    }
}

// ---------------------------------------------------------------------------
// Layer-0 embedding gathers (float4 granularity, 16 float4 per row).
// ---------------------------------------------------------------------------
__global__ void lgcn_gather0(const float* __restrict__ uemb,
                             const float* __restrict__ iemb,
                             const int*   __restrict__ users,
                             const int*   __restrict__ pos,
                             const int*   __restrict__ neg,
                             float* __restrict__ u0,
                             float* __restrict__ p0,
                             float* __restrict__ n0) {
    int idx = blockIdx.x * blockDim.x + threadIdx.x;             // float4 index
    if (idx >= BATCH * (DIM / 4)) return;
    int b = idx >> 4;
    int q = idx & 15;
    ((float4*)u0)[idx] = ((const float4*)uemb)[(long)users[b] * 16 + q];
    ((float4*)p0)[idx] = ((const float4*)iemb)[(long)pos[b]   * 16 + q];
    ((float4*)n0)[idx] = ((const float4*)iemb)[(long)neg[b]   * 16 + q];
}

// ---------------------------------------------------------------------------
extern "C" void kernel_launch(void* const* d_in, const int* in_sizes, int n_in,
                              void* d_out, int out_size, void* d_ws, size_t ws_size,
                              hipStream_t stream) {
    const float* uemb  = (const float*)d_in[0];
    const float* iemb  = (const float*)d_in[1];
    const float* vals  = (const float*)d_in[2];
    const int*   rows  = (const int*)d_in[3];
    const int*   cols  = (const int*)d_in[4];
    const int*   users = (const int*)d_in[5];
    const int*   pos   = (const int*)d_in[6];
    const int*   neg   = (const int*)d_in[7];
    const int    nnz   = in_sizes[3];

    float* out        = (float*)d_out;
    float* pos_scores = out;
    float* neg_scores = out + BATCH;
    float* u0         = out + 2 * BATCH;
    float* p0         = u0 + BATCH * DIM;
    float* n0         = p0 + BATCH * DIM;

    const size_t nfeat = (size_t)NNODES * DIM;
    float* acc  = (float*)d_ws;
    float* bufA = acc  + nfeat;
    float* bufB = bufA + nfeat;

    const long q4 = (long)nfeat / 4;
    lgcn_init<<<(int)((q4 + 255) / 256), 256, 0, stream>>>(uemb, iemb, acc, bufA, bufB);

    float* cur = bufA;
    float* nxt = bufB;
    for (int l = 0; l < NLAYERS; ++l) {
        int waves_per_block = 256 / 32;
        int grid = (nnz + waves_per_block - 1) / waves_per_block;
        lgcn_spmm<<<grid, 256, 0, stream>>>(rows, cols, vals, cur, nxt, nnz);
        lgcn_accum<<<(int)((q4 + 255) / 256), 256, 0, stream>>>(
            acc, nxt, cur, (l < NLAYERS - 1) ? 1 : 0);
        float* t = cur; cur = nxt; nxt = t;
    }

    // 512 tiles, 8 waves/block -> 64 blocks, exact coverage (EXEC all-ones).
    lgcn_score<<<(BATCH / 16) / 8, 256, 0, stream>>>(acc, users, pos, neg,
                                                     pos_scores, neg_scores);

    lgcn_gather0<<<(BATCH * (DIM / 4) + 255) / 256, 256, 0, stream>>>(
        uemb, iemb, users, pos, neg, u0, p0, n0);
}